// VADetector_37641093382267
// MI455X (gfx1250) — compile-verified
//
#include <hip/hip_runtime.h>

#define N_STATES  16
#define T_LEN     2048
#define BATCH     512
#define CHUNK     16

typedef __attribute__((ext_vector_type(2))) float v2f;
typedef __attribute__((ext_vector_type(8))) float v8f;

#if defined(__has_builtin)
#  if __has_builtin(__builtin_amdgcn_permlane16)
#    define VA_HAVE_PERMLANE16 1
#  endif
#endif

// Static intra-half permutation: dst lane i (within each 16-lane half) gets
// src lane sel_nibble[i].  old = 0 so the tied-dst init mov is independent of
// the loop-carried value (keeps it off the dependent chain).
__device__ __forceinline__ float va_perm16(float v, unsigned s1, unsigned s2,
                                           int fallback_src_lane) {
#ifdef VA_HAVE_PERMLANE16
    (void)fallback_src_lane;
    return __uint_as_float(__builtin_amdgcn_permlane16(
        0u, __float_as_uint(v), s1, s2, false, false));
#else
    (void)s1; (void)s2;
    return __shfl(v, fallback_src_lane, 32);
#endif
}

// Bare v_min_num_f32: skip fminf's sNaN canonicalization (v_max x,x on both
// operands) -- inputs are always ordinary finite values here.
__device__ __forceinline__ float va_min(float a, float b) {
#if defined(__gfx1250__)
    float r;
    asm("v_min_num_f32 %0, %1, %2" : "=v"(r) : "v"(a), "v"(b));
    return r;
#else
    return fminf(a, b);
#endif
}

// ---------------------------------------------------------------------------
// Forward ACS: one wave32 per batch element.
//   lane layout: s = lane & 15 (lanes 16..31 mirror the 16 states so EXEC is
//   all-ones for WMMA and permutes stay within each 16-lane half).
//   prior(t,s) = c*(y_t - e_s)^2 = [y, y^2, 1, 0] . [-2c*e, c, c*e^2, 0]
//   computed 16 timesteps at a time with V_WMMA_F32_16X16X4_F32.
// ---------------------------------------------------------------------------
__global__ __launch_bounds__(256) void va_forward_kernel(
    const float* __restrict__ y,          // [B, T]
    float* __restrict__ out_prob,         // [B, S, T]
    unsigned int* __restrict__ mask_ws)   // [B, T/2] packed decision masks
{
    const int lane = threadIdx.x & 31;
    const int wave = threadIdx.x >> 5;
    const int b    = blockIdx.x * (blockDim.x >> 5) + wave;
    const int s    = lane & 15;
    const bool hi  = lane >= 16;

    // expected[s] = sum_j (1 - 2*((s>>j)&1)) * exp(-0.2*j)
    const float h0 = 1.0f;
    const float h1 = 0.81873075307798182f;   // exp(-0.2)
    const float h2 = 0.67032004603563933f;   // exp(-0.4)
    const float h3 = 0.54881163609402639f;   // exp(-0.6)
    const float e  = ((s & 1) ? -h0 : h0) + ((s & 2) ? -h1 : h1)
                   + ((s & 4) ? -h2 : h2) + ((s & 8) ? -h3 : h3);
    const float c  = 5.0f;                   // 1 / (2 * sigma2), sigma2 = 0.1

    // Loop-invariant B matrix (4x16):
    //   VGPR0: lanes 0-15 -> row K0 = -2c*e_n ; lanes 16-31 -> row K2 = c*e_n^2
    //   VGPR1: lanes 0-15 -> row K1 = c       ; lanes 16-31 -> row K3 = 0
    v2f Bm;
    Bm[0] = hi ? (c * e * e) : (-2.0f * c * e);
    Bm[1] = hi ? 0.0f        : c;

    const float*  yb = y + (size_t)b * T_LEN;
    float*        ob = out_prob + ((size_t)b * N_STATES + s) * T_LEN;
    unsigned int* mb = mask_ws + (size_t)b * (T_LEN / 2);

    float p = 0.0f;                            // path metric for state s
    const int pred = (lane & 16) | (s >> 1);   // fallback bpermute source

    for (int t0 = 0; t0 < T_LEN; t0 += CHUNK) {
        float yv = yb[t0 + s];                 // halves read identical values
        // Unconditional, wrapped (always-valid) speculative prefetch.
        __builtin_prefetch(yb + ((t0 + CHUNK) & (T_LEN - 1)), 0, 3);

        // A matrix (16x4): VGPR0: K0 = y_t (lo) / K2 = 1 (hi)
        //                  VGPR1: K1 = y_t^2 (lo) / K3 = 0 (hi)
        v2f Am;
        Am[0] = hi ? 1.0f : yv;
        Am[1] = hi ? 0.0f : yv * yv;

        v8f D = {};
        D = __builtin_amdgcn_wmma_f32_16x16x4_f32(
                /*neg_a=*/false, Am, /*neg_b=*/false, Bm,
                /*c_mod=*/(short)0, D, /*reuse_a=*/false, /*reuse_b=*/false);
        // D[v]: lane<16 -> prior(t=v, s) ; lane>=16 -> prior(t=v+8, s)

        // Redistribute priors so every lane holds pri[t] for t = 0..15.
        float pri[CHUNK];
        #pragma unroll
        for (int i = 0; i < 8; ++i) {
            float dsw = __shfl_xor((float)D[i], 16, 32);   // cross-half copy
            pri[i]     = hi ? dsw        : (float)D[i];
            pri[i + 8] = hi ? (float)D[i] : dsw;
        }

        float obuf[CHUNK];
        unsigned int wreg = 0;     // lane k (k<8) collects mask word k of chunk
        unsigned int mlo  = 0;
        #pragma unroll
        for (int t = 0; t < CHUNK; ++t) {
            // ACS butterfly: cand_j = p[ j*8 + (s>>1) ]
            float p0 = va_perm16(p, 0x33221100u, 0x77665544u, pred);
            float p1 = va_perm16(p, 0xBBAA9988u, 0xFFEEDDCCu, pred | 8);
            bool  d1 = (p1 < p0);              // argmin picks idx 0 on ties
            p = va_min(p0, p1) + pri[t];       // 2-stage chain: min -> add
            obuf[t] = p;

            unsigned int m16 =
                (unsigned int)(__ballot(d1) & 0xFFFFu);    // uniform
            if ((t & 1) == 0) {
                mlo = m16;
            } else {
                unsigned int packed = mlo | (m16 << 16);
                wreg = (lane == (t >> 1)) ? packed : wreg; // deposit in lane t/2
            }
        }

        // One coalesced 32B mask store per chunk (8 lanes x 4B).
        if (lane < 8) mb[(t0 >> 1) + lane] = wreg;

        // Path-metric output: contiguous in t per (b,s), 16B-aligned stores.
        if (!hi) {
            float4* dst = (float4*)(ob + t0);
            #pragma unroll
            for (int q = 0; q < CHUNK / 4; ++q)
                dst[q] = make_float4(obuf[4*q+0], obuf[4*q+1],
                                     obuf[4*q+2], obuf[4*q+3]);
        }
    }
}

// ---------------------------------------------------------------------------
// Traceback: one thread per batch element, reverse scan over the stored
// decision masks. new_state = bit*8 + (state>>1); emitted bit == decision bit.
// ---------------------------------------------------------------------------
__global__ __launch_bounds__(256) void va_traceback_kernel(
    const unsigned int* __restrict__ mask_ws,   // [B, T/2]
    float* __restrict__ out_bits)               // [B, T]
{
    int b = blockIdx.x * blockDim.x + threadIdx.x;
    if (b >= BATCH) return;
    const unsigned int* mb = mask_ws + (size_t)b * (T_LEN / 2);
    float* ob = out_bits + (size_t)b * T_LEN;

    int state = 0;
    for (int t4 = T_LEN - 4; t4 >= 0; t4 -= 4) {
        unsigned int w1 = mb[(t4 >> 1) + 1];   // masks for t4+2 (lo), t4+3 (hi)
        unsigned int w0 = mb[(t4 >> 1) + 0];   // masks for t4+0 (lo), t4+1 (hi)
        float bits[4];
        unsigned int m;
        int bit;
        m = w1 >> 16;     bit = (m >> state) & 1; bits[3] = (float)bit; state = (bit << 3) | (state >> 1);
        m = w1 & 0xFFFFu; bit = (m >> state) & 1; bits[2] = (float)bit; state = (bit << 3) | (state >> 1);
        m = w0 >> 16;     bit = (m >> state) & 1; bits[1] = (float)bit; state = (bit << 3) | (state >> 1);
        m = w0 & 0xFFFFu; bit = (m >> state) & 1; bits[0] = (float)bit; state = (bit << 3) | (state >> 1);
        *(float4*)(ob + t4) = make_float4(bits[0], bits[1], bits[2], bits[3]);
    }
}

extern "C" void kernel_launch(void* const* d_in, const int* in_sizes, int n_in,
                              void* d_out, int out_size, void* d_ws, size_t ws_size,
                              hipStream_t stream) {
    (void)in_sizes; (void)n_in; (void)out_size; (void)ws_size;
    const float* y = (const float*)d_in[0];

    float* out_bits = (float*)d_out;                            // [B, T]
    float* out_prob = (float*)d_out + (size_t)BATCH * T_LEN;    // [B, S, T]
    unsigned int* mask_ws = (unsigned int*)d_ws;                // 2 MB scratch

    // 512 waves (one per batch), 8 waves per block -> 64 blocks.
    va_forward_kernel<<<BATCH / 8, 256, 0, stream>>>(y, out_prob, mask_ws);
    // 512 independent traceback chains.
    va_traceback_kernel<<<(BATCH + 255) / 256, 256, 0, stream>>>(mask_ws, out_bits);
}